// EnsembleModel_50044958933019
// MI455X (gfx1250) — compile-verified
//
#include <hip/hip_runtime.h>

typedef __attribute__((ext_vector_type(2))) float v2f;
typedef __attribute__((ext_vector_type(8))) float v8f;

#define NGROUPS 368
#define NMODELS 16
#define NBATCH  16384
#define GTILES  (NGROUPS / 16)   // 23
#define BPW     16               // batches per wave
#define WAVES_PER_BLOCK 8        // 256 threads = 8 wave32

// out[b, g0+i] = diag( X_tile(16x16) * W_tile^T(16x16) )[i] + bias[g0+i]
// computed as 4 chained V_WMMA_F32_16X16X4_F32 (K=16 total), fp32 throughout.
__global__ __launch_bounds__(256) void grouped_dot_wmma_f32(
    const float* __restrict__ x,    // [NBATCH, NGROUPS, NMODELS]
    const float* __restrict__ W,    // [NGROUPS, NMODELS]
    const float* __restrict__ bias, // [NGROUPS]
    float* __restrict__ out)        // [NBATCH, NGROUPS]
{
  const int lane = threadIdx.x & 31;
  const int wave = blockIdx.x * WAVES_PER_BLOCK + (threadIdx.x >> 5);
  const int total_waves = GTILES * (NBATCH / BPW);
  if (wave >= total_waves) return;   // wave-uniform

  const int gt     = wave % GTILES;
  const int bchunk = wave / GTILES;
  const int g0     = gt * 16;

  const int grp    = lane & 15;        // A row (M) / B col (N) owned by this lane
  const int hi     = lane >> 4;        // upper half-wave holds K = 2,3 of each slice
  const int rowoff = 2 * hi;           // element offset within a 16-float row

  // ---- B operand (weights), hoisted: Bm[t] feeds WMMA t (K = 4t .. 4t+3) ----
  // B vgpr v, lane L  ->  B[K = v + 2*hi, N = L&15] = W[g0 + (L&15), 4t + v + 2*hi]
  v2f Bm[4];
  const float* wrow = W + (g0 + grp) * NMODELS + rowoff;
  #pragma unroll
  for (int t = 0; t < 4; ++t)
    Bm[t] = *(const v2f*)(wrow + 4 * t);

  const float bv = bias[g0 + grp];

  const int xstride = NGROUPS * NMODELS;  // floats per batch row
  const float* xbase = x + (bchunk * BPW) * xstride + (g0 + grp) * NMODELS + rowoff;
  float* obase = out + (bchunk * BPW) * NGROUPS + g0 + grp;

  // Storing lanes: i<8 -> lane i (acc[i]); i>=8 -> lane i+16 (acc[i-8]).
  const bool storer = (lane < 8) || (lane >= 24);

  #pragma unroll 2
  for (int bi = 0; bi < BPW; ++bi) {
    const float* xp = xbase + bi * xstride;
    // A vgpr v, lane L -> A[M = L&15, K = v + 2*hi] = x[b, g0+(L&15), 4t + v + 2*hi]
    v2f A0 = __builtin_nontemporal_load((const v2f*)(xp + 0));
    v2f A1 = __builtin_nontemporal_load((const v2f*)(xp + 4));
    v2f A2 = __builtin_nontemporal_load((const v2f*)(xp + 8));
    v2f A3 = __builtin_nontemporal_load((const v2f*)(xp + 12));

    v8f acc = {};
    acc = __builtin_amdgcn_wmma_f32_16x16x4_f32(false, A0, false, Bm[0],
                                                (short)0, acc, false, false);
    acc = __builtin_amdgcn_wmma_f32_16x16x4_f32(false, A1, false, Bm[1],
                                                (short)0, acc, false, false);
    acc = __builtin_amdgcn_wmma_f32_16x16x4_f32(false, A2, false, Bm[2],
                                                (short)0, acc, false, false);
    acc = __builtin_amdgcn_wmma_f32_16x16x4_f32(false, A3, false, Bm[3],
                                                (short)0, acc, false, false);

    // Diagonal pick: each storing lane's value sits in its own acc[lane & 7].
    const int r = lane & 7;
    float s01 = (r & 1) ? acc[1] : acc[0];
    float s23 = (r & 1) ? acc[3] : acc[2];
    float s45 = (r & 1) ? acc[5] : acc[4];
    float s67 = (r & 1) ? acc[7] : acc[6];
    float s03 = (r & 2) ? s23 : s01;
    float s47 = (r & 2) ? s67 : s45;
    float res = (r & 4) ? s47 : s03;

    if (storer)
      __builtin_nontemporal_store(res + bv, obase + bi * NGROUPS);
  }
}

extern "C" void kernel_launch(void* const* d_in, const int* in_sizes, int n_in,
                              void* d_out, int out_size, void* d_ws, size_t ws_size,
                              hipStream_t stream) {
  const float* x    = (const float*)d_in[0];
  const float* W    = (const float*)d_in[1];
  const float* bias = (const float*)d_in[2];
  float* out        = (float*)d_out;

  const int total_waves = GTILES * (NBATCH / BPW);            // 23552
  const int blocks = (total_waves + WAVES_PER_BLOCK - 1) / WAVES_PER_BLOCK; // 2944
  grouped_dot_wmma_f32<<<blocks, 256, 0, stream>>>(x, W, bias, out);
}